// STFT_54065048322275
// MI455X (gfx1250) — compile-verified
//
#include <hip/hip_runtime.h>
#include <math.h>

#define FILTER_LENGTH 1024
#define HOP           512
#define CUTOFF        513      // FILTER_LENGTH/2 + 1
#define NROWS         1026     // 2*CUTOFF rows in forward_basis
#define T_SAMPLES     1048576
#define N_FRAMES      2049

#define KC     32              // K-chunk per LDS stage
#define KCP    36              // padded LDS row stride (floats): conflict-free + 16B aligned
#define F_TILE 64              // freqs per block (real) ; imag rows staged alongside
#define N_TILE 32              // frames per block
#define NKC    (FILTER_LENGTH / KC)

typedef float v2f __attribute__((ext_vector_type(2)));
typedef float v8f __attribute__((ext_vector_type(8)));

// ---- CDNA5 async global->LDS path (guarded; falls back to VGPR staging) ----
#if defined(__AMDGCN__) && \
    __has_builtin(__builtin_amdgcn_global_load_async_to_lds_b128) && \
    __has_builtin(__builtin_amdgcn_global_load_async_to_lds_b32)
#define USE_ASYNC_LDS 1
#else
#define USE_ASYNC_LDS 0
#endif

#define AS_GLOBAL __attribute__((address_space(1)))
#define AS_LOCAL  __attribute__((address_space(3)))

// pointee types per the builtin signatures (width encoded in pointee)
typedef int v4i_t __attribute__((vector_size(16)));

#if USE_ASYNC_LDS
static __device__ __forceinline__ void wait_async0() {
#if __has_builtin(__builtin_amdgcn_s_wait_asynccnt)
    __builtin_amdgcn_s_wait_asynccnt(0);
#else
    asm volatile("s_wait_asynccnt 0x0" ::: "memory");
#endif
}
#endif

__global__ __launch_bounds__(256, 2)
void stft_wmma_f32_kernel(const float* __restrict__ x,
                          const float* __restrict__ fb,
                          const float* __restrict__ eps_p,
                          float* __restrict__ out)
{
    // double-buffered LDS: 2*(128*36 + 32*36)*4B = 46,080 B
    __shared__ __align__(16) float lds_fb [2][128 * KCP];
    __shared__ __align__(16) float lds_sig[2][N_TILE * KCP];

    const int tid = threadIdx.x;
    const int b   = blockIdx.z;
    const int f0  = blockIdx.y * F_TILE;
    const int n0  = blockIdx.x * N_TILE;

    const float* __restrict__ xb = x + (size_t)b * T_SAMPLES;

    // stage one K-chunk [k0, k0+KC) into buffer `buf`
    auto stage = [&](int buf, int k0) {
        // FB tile: 128 rows x KC (rows 0..63 real = f0+r ; 64..127 imag = CUTOFF+f0+(r-64))
        #pragma unroll
        for (int i = tid; i < 128 * (KC / 4); i += 256) {
            int r  = i >> 3;           // / (KC/4)
            int c4 = (i & 7) << 2;     // *4
            int gr = (r < 64) ? (f0 + r) : (CUTOFF + f0 + (r - 64));
            gr = (gr > NROWS - 1) ? (NROWS - 1) : gr;   // clamp ragged tail in-bounds
            const float* src = fb + (size_t)gr * FILTER_LENGTH + k0 + c4;
            float*       dst = &lds_fb[buf][r * KCP + c4];
#if USE_ASYNC_LDS
            __builtin_amdgcn_global_load_async_to_lds_b128(
                (AS_GLOBAL v4i_t*)src, (AS_LOCAL v4i_t*)dst, 0, 0);
#else
            *(float4*)dst = *(const float4*)src;        // row stride 144B -> 16B aligned
#endif
        }
        // signal tile: 32 frames x KC, reflect-padded gather (each frame chunk contiguous)
        #pragma unroll
        for (int i = tid; i < N_TILE * KC; i += 256) {
            int  fr = i >> 5;
            int  c  = i & 31;
            long j  = (long)(n0 + fr) * HOP + k0 + c - (FILTER_LENGTH / 2);
            j = (j < 0) ? -j : j;
            j = (j >= T_SAMPLES) ? (2L * T_SAMPLES - 2 - j) : j;
#if USE_ASYNC_LDS
            __builtin_amdgcn_global_load_async_to_lds_b32(
                (AS_GLOBAL int*)(xb + j),
                (AS_LOCAL int*)&lds_sig[buf][fr * KCP + c], 0, 0);
#else
            lds_sig[buf][fr * KCP + c] = xb[j];
#endif
        }
    };

    const int wave = tid >> 5;
    const int lane = tid & 31;
    const int half = lane >> 4;   // K-pair selector per WMMA layout
    const int m    = lane & 15;   // M (for A) / N (for B) within 16x16 tile
    const int fsub = wave & 3;    // 4 freq sub-tiles
    const int nsub = wave >> 2;   // 2 frame sub-tiles

    v8f accR = {};
    v8f accI = {};

    stage(0, 0);

    #pragma unroll 1
    for (int kc = 0; kc < NKC; ++kc) {
        const int buf = kc & 1;
#if USE_ASYNC_LDS
        wait_async0();                 // producer: my async LDS writes done
#endif
        __syncthreads();               // consumer ds reads drained by syncthreads (dscnt)
        if (kc + 1 < NKC) stage(buf ^ 1, (kc + 1) * KC); // prefetch next chunk

        const float* Ar = &lds_fb [buf][(fsub * 16 + m)      * KCP + 2 * half];
        const float* Ai = &lds_fb [buf][(64 + fsub * 16 + m) * KCP + 2 * half];
        const float* Bp = &lds_sig[buf][(nsub * 16 + m)      * KCP + 2 * half];

        #pragma unroll
        for (int s = 0; s < KC / 4; ++s) {             // 8 x (2 WMMA) per chunk
            v2f ar = *(const v2f*)(Ar + s * 4);
            v2f ai = *(const v2f*)(Ai + s * 4);
            v2f bb = *(const v2f*)(Bp + s * 4);
            accR = __builtin_amdgcn_wmma_f32_16x16x4_f32(
                       false, ar, false, bb, (short)0, accR, false, false);
            accI = __builtin_amdgcn_wmma_f32_16x16x4_f32(
                       false, ai, false, bb, (short)0, accI, false, false);
        }
    }

    // epilogue: magnitude, masked ragged edges
    const float eps   = *eps_p;
    const int   fbase = f0 + fsub * 16 + 8 * half;
    const int   n     = n0 + nsub * 16 + m;
    if (n < N_FRAMES) {
        #pragma unroll
        for (int r = 0; r < 8; ++r) {
            const int f = fbase + r;                   // M = r + 8*(lane/16)
            if (f < CUTOFF) {
                const float vr = accR[r];
                const float vi = accI[r];
                out[((size_t)b * CUTOFF + f) * N_FRAMES + n] =
                    sqrtf(vr * vr + vi * vi + eps);
            }
        }
    }
}

extern "C" void kernel_launch(void* const* d_in, const int* in_sizes, int n_in,
                              void* d_out, int out_size, void* d_ws, size_t ws_size,
                              hipStream_t stream)
{
    const float* x   = (const float*)d_in[0];  // (B, 1048576) f32
    const float* fb  = (const float*)d_in[1];  // (1026, 1024) f32
    const float* eps = (const float*)d_in[2];  // scalar f32
    float* out = (float*)d_out;                // (B, 513, 2049) f32

    const int B = in_sizes[0] / T_SAMPLES;

    dim3 grid((N_FRAMES + N_TILE - 1) / N_TILE,   // 65
              (CUTOFF   + F_TILE - 1) / F_TILE,   // 9
              B);                                  // 8
    stft_wmma_f32_kernel<<<grid, dim3(256), 0, stream>>>(x, fb, eps, out);
}